// GATConv_18391049961913
// MI455X (gfx1250) — compile-verified
//
#include <hip/hip_runtime.h>

#define NEG_SLOPE 0.2f

typedef __attribute__((ext_vector_type(2))) float v2f;
typedef __attribute__((ext_vector_type(8))) float v8f;
typedef int v4i_vs __attribute__((vector_size(4 * sizeof(int))));

#define LDS_AS __attribute__((address_space(3)))
#define GLB_AS __attribute__((address_space(1)))

// ---------------------------------------------------------------------------
// K1: h = x @ W  via V_WMMA_F32_16X16X4_F32 (full fp32 precision)
//   grid.x = N/16 node tiles, 256 threads = 8 wave32
//   x tile (16x128 fp32 = 8KB) staged to LDS via async-load, shared by 8 waves
//   wave w computes the 16x16 tile for head/col-tile w (OUT_CH == 16 == WMMA N)
// ---------------------------------------------------------------------------
__global__ __launch_bounds__(256) void gat_gemm_wmma(
    const float* __restrict__ x,   // (N,128)
    const float* __restrict__ W,   // (128,128)
    float* __restrict__ h,         // (N,128)
    int N) {
  __shared__ float sx[16 * 128];   // 8 KB x-tile

  const int tid  = threadIdx.x;
  const int wave = tid >> 5;           // 0..7 -> column tile (== head)
  const int lane = tid & 31;
  const int hi   = lane >> 4;          // 0/1: K sub-pair / M half
  const int lo   = lane & 15;
  const int rowbase = blockIdx.x * 16;
  const int colbase = wave * 16;
  const bool fulltile = (rowbase + 16 <= N);   // always true when N%16==0

  if (fulltile) {
    // Stage the contiguous 8KB tile: 256 threads x 32B (2x B128 async each)
    const char* gsrc  = (const char*)(x + (size_t)rowbase * 128) + tid * 32;
    char*       lbase = (char*)sx + tid * 32;
#if __has_builtin(__builtin_amdgcn_global_load_async_to_lds_b128)
    __builtin_amdgcn_global_load_async_to_lds_b128(
        (GLB_AS v4i_vs*)gsrc, (LDS_AS v4i_vs*)lbase, 0, 0);
    __builtin_amdgcn_global_load_async_to_lds_b128(
        (GLB_AS v4i_vs*)(gsrc + 16), (LDS_AS v4i_vs*)(lbase + 16), 0, 0);
#if __has_builtin(__builtin_amdgcn_s_wait_asynccnt)
    __builtin_amdgcn_s_wait_asynccnt(0);
#else
    asm volatile("s_wait_asynccnt 0x0" ::: "memory");
#endif
#else
    // Fallback: plain cooperative copy (global_load_b128 + ds_store_b128)
    const float4* gs = (const float4*)gsrc;
    float4*       ls = (float4*)lbase;
    ls[0] = gs[0];
    ls[1] = gs[1];
#endif
  } else {
    // Ragged tail (unused for N%16==0): guarded copy with clamped rows
    for (int idx = tid; idx < 16 * 128; idx += 256) {
      int rr = rowbase + (idx >> 7);
      if (rr >= N) rr = N - 1;
      sx[idx] = x[(size_t)rr * 128 + (idx & 127)];
    }
  }
  __syncthreads();

  const float* xrow = sx + lo * 128;   // LDS row for this lane's A fragment

  v8f c = {};
  #pragma unroll 4
  for (int k0 = 0; k0 < 128; k0 += 4) {
    const int k = k0 + 2 * hi;
    // A 16x4 fp32: v0 = K=k0 (lanes 0-15) / K=k0+2 (lanes 16-31), v1 = +1
    v2f a = *(const v2f*)(xrow + k);             // ds_load_b64 (8B aligned)
    // B 4x16 fp32: v0 = row k0 (lanes 0-15) / row k0+2 (lanes 16-31), v1 = +1
    v2f b;
    b.x = W[(size_t)k * 128 + colbase + lo];
    b.y = W[(size_t)(k + 1) * 128 + colbase + lo];
    c = __builtin_amdgcn_wmma_f32_16x16x4_f32(
        /*neg_a=*/false, a, /*neg_b=*/false, b,
        /*c_mod=*/(short)0, c, /*reuse_a=*/false, /*reuse_b=*/false);
  }

  // C/D layout: VGPR r, lanes 0-15 -> M=r; lanes 16-31 -> M=r+8; N=lo
  float* hp = h + ((size_t)(rowbase + 8 * hi)) * 128 + colbase + lo;
  if (fulltile) {
    #pragma unroll
    for (int r = 0; r < 8; ++r) hp[(size_t)r * 128] = c[r];
  } else {
    #pragma unroll
    for (int r = 0; r < 8; ++r) {
      if (rowbase + r + 8 * hi < N) hp[(size_t)r * 128] = c[r];
    }
  }
}

// ---------------------------------------------------------------------------
// K2: per-(node,head): a_src/a_dst dots; init emax=-inf, denom=0, out=bias
// ---------------------------------------------------------------------------
__global__ void gat_node(
    const float* __restrict__ h,
    const float* __restrict__ att_src,  // (8,16)
    const float* __restrict__ att_dst,  // (8,16)
    const float* __restrict__ bias,     // (128,)
    float* __restrict__ a_src, float* __restrict__ a_dst,
    unsigned* __restrict__ emax_bits, float* __restrict__ denom,
    float* __restrict__ out, int n_items) {
  const int t = blockIdx.x * blockDim.x + threadIdx.x;
  if (t >= n_items) return;
  const int n = t >> 3, hh = t & 7;
  const float* hp = h + (size_t)n * 128 + hh * 16;
  const float* as = att_src + hh * 16;
  const float* ad = att_dst + hh * 16;
  float s = 0.f, d = 0.f;
  #pragma unroll
  for (int cidx = 0; cidx < 16; ++cidx) {
    const float v = hp[cidx];
    s = fmaf(v, as[cidx], s);
    d = fmaf(v, ad[cidx], d);
  }
  a_src[t] = s;
  a_dst[t] = d;
  emax_bits[t] = 0xFF800000u;  // -inf
  denom[t] = 0.f;
  float* op = out + (size_t)n * 128 + hh * 16;
  const float* bp = bias + hh * 16;
  #pragma unroll
  for (int cidx = 0; cidx < 16; ++cidx) op[cidx] = bp[cidx];
}

// Ordered-int float atomic max (valid for all finite floats, -inf init).
__device__ __forceinline__ void atomic_max_float(unsigned* addr, float v) {
  if (v >= 0.f) {
    atomicMax((int*)addr, (int)__float_as_uint(v == 0.f ? 0.f : v));
  } else {
    atomicMin(addr, __float_as_uint(v));
  }
}

__device__ __forceinline__ float leaky(float e) {
  return e > 0.f ? e : e * NEG_SLOPE;
}

// ---------------------------------------------------------------------------
// K3: segment max over incoming edges (self-loops = items [E, E+N))
// ---------------------------------------------------------------------------
__global__ void gat_edge_max(
    const int* __restrict__ src, const int* __restrict__ dst,
    const float* __restrict__ a_src, const float* __restrict__ a_dst,
    unsigned* __restrict__ emax_bits, int E, int n_items) {
  const int t = blockIdx.x * blockDim.x + threadIdx.x;
  if (t >= n_items) return;
  const int e = t >> 3, hh = t & 7;
  int j, i;
  if (e < E) { j = src[e]; i = dst[e]; } else { j = i = e - E; }
  const float lg = leaky(a_src[j * 8 + hh] + a_dst[i * 8 + hh]);
  atomic_max_float(&emax_bits[i * 8 + hh], lg);
}

// ---------------------------------------------------------------------------
// K4: segment sum of exp(e - emax)
// ---------------------------------------------------------------------------
__global__ void gat_edge_sum(
    const int* __restrict__ src, const int* __restrict__ dst,
    const float* __restrict__ a_src, const float* __restrict__ a_dst,
    const unsigned* __restrict__ emax_bits, float* __restrict__ denom,
    int E, int n_items) {
  const int t = blockIdx.x * blockDim.x + threadIdx.x;
  if (t >= n_items) return;
  const int e = t >> 3, hh = t & 7;
  int j, i;
  if (e < E) { j = src[e]; i = dst[e]; } else { j = i = e - E; }
  const float lg = leaky(a_src[j * 8 + hh] + a_dst[i * 8 + hh]);
  const float m = __uint_as_float(emax_bits[i * 8 + hh]);
  atomicAdd(&denom[i * 8 + hh], __expf(lg - m));
}

// ---------------------------------------------------------------------------
// K5: out[i, hh*16..] += alpha * h[j, hh*16..]  (one thread per (edge,head))
// ---------------------------------------------------------------------------
__global__ void gat_edge_agg(
    const int* __restrict__ src, const int* __restrict__ dst,
    const float* __restrict__ a_src, const float* __restrict__ a_dst,
    const unsigned* __restrict__ emax_bits, const float* __restrict__ denom,
    const float* __restrict__ h, float* __restrict__ out,
    int E, int n_items) {
  const int t = blockIdx.x * blockDim.x + threadIdx.x;
  if (t >= n_items) return;
  const int e = t >> 3, hh = t & 7;
  int j, i;
  if (e < E) { j = src[e]; i = dst[e]; } else { j = i = e - E; }
  const float lg = leaky(a_src[j * 8 + hh] + a_dst[i * 8 + hh]);
  const float m = __uint_as_float(emax_bits[i * 8 + hh]);
  const float alpha = __expf(lg - m) / denom[i * 8 + hh];

  const float4* hp = (const float4*)(h + (size_t)j * 128 + hh * 16);
  float* op = out + (size_t)i * 128 + hh * 16;
  #pragma unroll
  for (int q = 0; q < 4; ++q) {
    const float4 v = hp[q];
    atomicAdd(op + 4 * q + 0, alpha * v.x);
    atomicAdd(op + 4 * q + 1, alpha * v.y);
    atomicAdd(op + 4 * q + 2, alpha * v.z);
    atomicAdd(op + 4 * q + 3, alpha * v.w);
  }
}

// ---------------------------------------------------------------------------
extern "C" void kernel_launch(void* const* d_in, const int* in_sizes, int n_in,
                              void* d_out, int out_size, void* d_ws, size_t ws_size,
                              hipStream_t stream) {
  const float* x        = (const float*)d_in[0];  // (N,128)
  const int*   edge     = (const int*)d_in[1];    // (2,E)
  const float* W        = (const float*)d_in[2];  // (128,128)
  const float* att_src  = (const float*)d_in[3];  // (8,16)
  const float* att_dst  = (const float*)d_in[4];  // (8,16)
  const float* bias     = (const float*)d_in[5];  // (128,)
  float*       out      = (float*)d_out;

  const int N = in_sizes[0] / 128;
  const int E = in_sizes[1] / 2;
  const int* srcp = edge;
  const int* dstp = edge + E;

  // workspace partition
  char* ws = (char*)d_ws;
  float*    h     = (float*)ws;    ws += (size_t)N * 128 * sizeof(float);
  float*    asrc  = (float*)ws;    ws += (size_t)N * 8 * sizeof(float);
  float*    adst  = (float*)ws;    ws += (size_t)N * 8 * sizeof(float);
  unsigned* emax  = (unsigned*)ws; ws += (size_t)N * 8 * sizeof(unsigned);
  float*    denom = (float*)ws;

  // K1: GEMM (WMMA f32 + async LDS staging)
  gat_gemm_wmma<<<(N + 15) / 16, 256, 0, stream>>>(x, W, h, N);

  // K2: attention coefficients + init
  const int nh = N * 8;
  gat_node<<<(nh + 255) / 256, 256, 0, stream>>>(
      h, att_src, att_dst, bias, asrc, adst, emax, denom, out, nh);

  // K3/K4/K5: edge passes (self-loops appended as virtual edges)
  const int items = (E + N) * 8;
  const int blocks = (items + 255) / 256;
  gat_edge_max<<<blocks, 256, 0, stream>>>(srcp, dstp, asrc, adst, emax, E, items);
  gat_edge_sum<<<blocks, 256, 0, stream>>>(srcp, dstp, asrc, adst, emax, denom, E, items);
  gat_edge_agg<<<blocks, 256, 0, stream>>>(srcp, dstp, asrc, adst, emax, denom, h, out, E, items);
}